// LocallyConnected2d_541165879368
// MI455X (gfx1250) — compile-verified
//
#include <hip/hip_runtime.h>

typedef __attribute__((ext_vector_type(2))) float v2f;
typedef __attribute__((ext_vector_type(8))) float v8f;

#define NBATCH 32
#define CIN    1024
#define COUT   256
#define HW     7
#define NP     49            // 7*7 spatial positions
#define K9     9
#define CC     8             // input channels per K-chunk
#define KC     (CC * K9)     // 72 K-steps per chunk
#define LDP    (KC + 4)      // 76: bank stride 12 mod 64 -> conflict-free
#define NCHUNK (CIN / CC)    // 128

#if __has_builtin(__builtin_amdgcn_global_load_async_to_lds_b32)
#define HAVE_ASYNC_LDS 1
#else
#define HAVE_ASYNC_LDS 0
#endif

typedef __attribute__((address_space(1))) int GInt;  // global int
typedef __attribute__((address_space(3))) int LInt;  // LDS int

__device__ __forceinline__ void cp_async_b32(const float* g, float* l) {
#if HAVE_ASYNC_LDS
  __builtin_amdgcn_global_load_async_to_lds_b32((GInt*)g, (LInt*)l, 0, 0);
#else
  *l = *g;   // synchronous fallback
#endif
}

__device__ __forceinline__ void async_wait_all() {
#if HAVE_ASYNC_LDS
#if __has_builtin(__builtin_amdgcn_s_wait_asynccnt)
  __builtin_amdgcn_s_wait_asynccnt(0);
#else
  asm volatile("s_wait_asynccnt 0" ::: "memory");
#endif
#endif
}

__launch_bounds__(128)
__global__ void lc2d_wmma_async(const float* __restrict__ x,
                                const float* __restrict__ w,
                                const float* __restrict__ bias,
                                float* __restrict__ out) {
  // double-buffered staging: 2 * (32 + 32) * 76 * 4B = 38,912 B
  __shared__ float Alds[2][NBATCH][LDP];  // [buf][batch][K]   : x patches
  __shared__ float Blds[2][32][LDP];      // [buf][o_local][K] : weights

  const int p     = blockIdx.x >> 3;      // 0..48 spatial position
  const int cg    = blockIdx.x & 7;       // output-channel group of 32
  const int oh    = p / HW, ow = p % HW;
  const int obase = cg * 32;

  const int tid   = threadIdx.x;
  const int lane  = tid & 31;
  const int wave  = tid >> 5;             // 0..3
  const int mt    = wave & 1;             // batch tile: rows 16*mt..16*mt+15
  const int nt    = wave >> 1;            // channel sub-tile of 16
  const int l16   = lane & 15;
  const int khalf = (lane >> 4) << 1;     // K offset 0 / 2 per half-wave

  v8f acc = {0.f, 0.f, 0.f, 0.f, 0.f, 0.f, 0.f, 0.f};

  // ---- fill helpers ----
  auto fillA = [&](int buf, int c0) {     // zero-padded 3x3 patches (L2-resident)
    for (int e = tid; e < NBATCH * KC; e += 128) {
      int n  = e / KC;
      int r  = e % KC;
      int cl = r / K9;
      int k  = r % K9;
      int hh = oh + (k / 3) - 1;
      int ww = ow + (k % 3) - 1;
      float v = 0.f;
      if ((unsigned)hh < HW && (unsigned)ww < HW)
        v = x[(((size_t)n * CIN + (size_t)(c0 + cl)) * HW + hh) * HW + ww];
      Alds[buf][n][r] = v;
    }
  };
  auto fillB = [&](int buf, int c0) {     // 462MB weight stream: async -> LDS
    for (int e = tid; e < 32 * KC; e += 128) {
      int ol = e / KC;
      int r  = e % KC;
      int cl = r / K9;
      int k  = r % K9;
      size_t gi = ((((size_t)(obase + ol) * CIN + (size_t)(c0 + cl)) * NP) + p) * K9 + k;
      cp_async_b32(&w[gi], &Blds[buf][ol][r]);
    }
  };

  // ---- prologue: stage chunk 0 ----
  fillB(0, 0);
  fillA(0, 0);

  for (int cc = 0; cc < NCHUNK; ++cc) {
    const int cur = cc & 1;
    const int nxt = cur ^ 1;

    async_wait_all();       // my async fills for chunk cc are in LDS
    __syncthreads();        // everyone's fills visible; prior reads of 'nxt' done

    if (cc + 1 < NCHUNK) {  // overlap next chunk's stream with this chunk's WMMAs
      fillB(nxt, (cc + 1) * CC);
      fillA(nxt, (cc + 1) * CC);
    }

    const float* arow = &Alds[cur][mt * 16 + l16][0];
    const float* brow = &Blds[cur][nt * 16 + l16][0];
#pragma unroll
    for (int kk = 0; kk < KC; kk += 4) {
      v2f a = *(const v2f*)(arow + kk + khalf);
      v2f b = *(const v2f*)(brow + kk + khalf);
      acc = __builtin_amdgcn_wmma_f32_16x16x4_f32(false, a, false, b,
                                                  (short)0, acc, false, false);
    }
  }

  // ---- epilogue: bias + LeakyReLU(0.1), stores strided by NP across lanes ----
  const int   o   = obase + nt * 16 + l16;
  const float bv  = bias[o * NP + p];
  const int   nb0 = mt * 16 + ((lane >> 4) << 3);  // lanes 16-31 hold M+8
#pragma unroll
  for (int r = 0; r < 8; ++r) {
    float v = acc[r] + bv;
    v = (v >= 0.f) ? v : 0.1f * v;
    out[((size_t)(nb0 + r) * COUT + o) * NP + p] = v;
  }
}

extern "C" void kernel_launch(void* const* d_in, const int* in_sizes, int n_in,
                              void* d_out, int out_size, void* d_ws, size_t ws_size,
                              hipStream_t stream) {
  const float* x    = (const float*)d_in[0];
  const float* wgt  = (const float*)d_in[1];
  const float* bias = (const float*)d_in[2];
  float* out        = (float*)d_out;

  dim3 grid(NP * (COUT / 32));  // 49 * 8 = 392 blocks
  dim3 block(128);              // 4 wave32s
  hipLaunchKernelGGL(lc2d_wmma_async, grid, block, 0, stream, x, wgt, bias, out);
}